// DetectionConfidenceMap2keypoint_test_39926015984323
// MI455X (gfx1250) — compile-verified
//
#include <hip/hip_runtime.h>
#include <math.h>

// Problem constants (from the reference: [32, 128, 96, 96] f32)
#define BATCH   32
#define CHN     128
#define HT      96
#define WD      96
#define HW      (HT * WD)          // 9216 pixels per plane
#define TPB     256                // threads per block = 8 waves of 32
#define CHUNK   16                 // channels staged per WMMA reduction round
#define PAD     17                 // LDS row stride (floats): gcd(17,64)=1 -> conflict-free

typedef __attribute__((ext_vector_type(2))) float v2f;   // 16x4 f32 A/B frag = 2 VGPRs
typedef __attribute__((ext_vector_type(8))) float v8f;   // 16x16 f32 C/D frag = 8 VGPRs

// ---------------------------------------------------------------------------
// Zero the global accumulators (zeta / kpx / kpy, 3 * B*C floats in d_ws)
// ---------------------------------------------------------------------------
__global__ void init_acc_kernel(float* __restrict__ acc, int n) {
    int i = blockIdx.x * blockDim.x + threadIdx.x;
    if (i < n) acc[i] = 0.0f;
}

// ---------------------------------------------------------------------------
// Fused channel-softmax + coordinate reductions.
// grid = (HW/TPB, BATCH); block = TPB. Each thread owns one pixel p and walks
// all 128 channels (coalesced: consecutive lanes -> consecutive w; channel
// stride fits the 24-bit immediate so it is one base pointer + immediates).
// Per 16-channel chunk: softmax written to d_out, values staged in LDS, then
// v_wmma_f32_16x16x4_f32 computes  D[q][ch] = sum_p coef_q(p) * S[p][ch]
// with coef rows {x(p), y(p)} (softmax tile) and {1} (raw tile -> zeta).
// Each wave's 32 pixels lie in ONE image row (96 = 3*32), so y is a wave
// constant and x advances by +4.0f per K-step: A fragments are built once
// and updated with a single v_add_f32 per element.
// ---------------------------------------------------------------------------
__global__ __launch_bounds__(TPB)
void softmax_kp_kernel(const float* __restrict__ in, float* __restrict__ out,
                       float* __restrict__ gz, float* __restrict__ gx,
                       float* __restrict__ gy) {
    __shared__ float s_tile[TPB * PAD];   // softmax staging  [pixel][chan]
    __shared__ float r_tile[TPB * PAD];   // raw-value staging
    __shared__ float az[CHN];             // per-block zeta  accumulators
    __shared__ float ax[CHN];             // per-block kp_x  accumulators
    __shared__ float ay[CHN];             // per-block kp_y  accumulators

    const int tid = threadIdx.x;
    const int b   = blockIdx.y;
    const int p0  = blockIdx.x * TPB;     // block pixel base
    const int p   = p0 + tid;             // this thread's pixel (0..HW-1)

    const float* __restrict__ base  = in  + (size_t)b * CHN * HW + p;
    float*       __restrict__ obase = out + (size_t)b * CHN * HW + p;

    if (tid < CHN) { az[tid] = 0.0f; ax[tid] = 0.0f; ay[tid] = 0.0f; }

    // ---- pass 1: channel max (numerically stable softmax) ----
    float m = -INFINITY;
#pragma unroll 8
    for (int c = 0; c < CHN; ++c) m = fmaxf(m, base[c * HW]);

    // ---- pass 2: sum of exp (re-reads hit L2/L0) ----
    float sum = 0.0f;
#pragma unroll 8
    for (int c = 0; c < CHN; ++c) sum += __expf(base[c * HW] - m);
    const float inv = 1.0f / sum;

    __syncthreads();                      // accumulator init visible

    const int wave  = tid >> 5;           // 8 waves per block
    const int lane  = tid & 31;
    const int pw    = wave << 5;          // wave's pixel base within block
    const int mrow  = lane & 15;          // WMMA A row (M) this lane feeds
    const int khalf = (lane >> 4) << 1;   // lanes 0-15: K=0/1, lanes 16-31: K=2/3
    const int ch    = lane & 15;          // WMMA B column (N) this lane feeds

    // Wave-constant A-fragment ingredients (one row per wave).
    const int   wbase = p0 + pw;                       // multiple of 32
    const float yf    = (float)(wbase / WD);           // row index (constant)
    const float xb    = (float)(wbase % WD + khalf);   // x of this lane's K slot 0
    v2f a1_init, a2;
    a1_init[0] = (mrow == 0) ? xb          : ((mrow == 1) ? yf : 0.0f);
    a1_init[1] = (mrow == 0) ? (xb + 1.0f) : ((mrow == 1) ? yf : 0.0f);
    a2[0]      = (mrow == 0) ? 1.0f : 0.0f;            // ones row -> zeta
    a2[1]      = a2[0];
    const float dx = (mrow == 0) ? 4.0f : 0.0f;        // x advance per K-step

    for (int chunk = 0; chunk < CHN / CHUNK; ++chunk) {
        // ---- pass 3 (chunked): softmax out + LDS staging ----
#pragma unroll
        for (int i = 0; i < CHUNK; ++i) {
            const int c = chunk * CHUNK + i;
            const float v = base[c * HW];
            const float s = __expf(v - m) * inv;
            obase[c * HW] = s;                    // map_val_all
            s_tile[tid * PAD + i] = s;
            r_tile[tid * PAD + i] = v;
        }
        __syncthreads();

        // ---- WMMA reduction over this wave's 32 pixels (8 K-steps of 4) ----
        v8f acc1 = {};        // rows: 0 -> sum s*x, 1 -> sum s*y
        v8f acc2 = {};        // row 0 -> sum raw (zeta)
        v2f a1 = a1_init;
#pragma unroll
        for (int j = 0; j < 8; ++j) {
            const int pk0 = pw + 4 * j + khalf;   // pixel for this lane's K slot 0
            const int pk1 = pk0 + 1;              // ... K slot 1

            // B fragments from LDS (4x16 f32, rows striped like A's K pattern)
            v2f b1, b2;
            b1[0] = s_tile[pk0 * PAD + ch];
            b1[1] = s_tile[pk1 * PAD + ch];
            b2[0] = r_tile[pk0 * PAD + ch];
            b2[1] = r_tile[pk1 * PAD + ch];

            acc1 = __builtin_amdgcn_wmma_f32_16x16x4_f32(
                false, a1, false, b1, (short)0, acc1, false, false);
            acc2 = __builtin_amdgcn_wmma_f32_16x16x4_f32(
                false, a2, false, b2, (short)0, acc2, false, false);

            a1[0] += dx;                          // advance x coords (y rows: +0)
            a1[1] += dx;
        }

        // D rows 0..7 live in VGPR 0..7 of lanes 0-15 (cols = lane)
        if (lane < 16) {
            const int c = chunk * CHUNK + lane;
            atomicAdd(&ax[c], acc1[0]);   // row 0: sum s*x
            atomicAdd(&ay[c], acc1[1]);   // row 1: sum s*y
            atomicAdd(&az[c], acc2[0]);   // row 0: sum raw
        }
        __syncthreads();                  // tiles reusable next chunk
    }

    // ---- block -> global accumulation ----
    if (tid < CHN) {
        const int g = b * CHN + tid;
        atomicAdd(&gz[g], az[tid]);
        atomicAdd(&gx[g], ax[tid]);
        atomicAdd(&gy[g], ay[tid]);
    }
}

// ---------------------------------------------------------------------------
// keypoint[b,c,:] = {rint(kpx/zeta), rint(kpy/zeta)}; also emit zeta.
// rintf == round-half-to-even, matching jnp.round.
// ---------------------------------------------------------------------------
__global__ void finalize_kernel(const float* __restrict__ gz,
                                const float* __restrict__ gx,
                                const float* __restrict__ gy,
                                float* __restrict__ out_kp,
                                float* __restrict__ out_zeta) {
    int i = blockIdx.x * blockDim.x + threadIdx.x;
    if (i < BATCH * CHN) {
        const float z = gz[i];
        out_zeta[i]       = z;
        out_kp[2 * i + 0] = rintf(gx[i] / z);
        out_kp[2 * i + 1] = rintf(gy[i] / z);
    }
}

// ---------------------------------------------------------------------------
extern "C" void kernel_launch(void* const* d_in, const int* in_sizes, int n_in,
                              void* d_out, int out_size, void* d_ws, size_t ws_size,
                              hipStream_t stream) {
    (void)in_sizes; (void)n_in; (void)out_size; (void)ws_size;

    const float* in = (const float*)d_in[0];   // [32,128,96,96] f32
    // d_in[1] = cur_batch scalar (device) — shapes are fixed, ignore.

    float* out      = (float*)d_out;
    float* out_map  = out;                                    // 37,748,736
    float* out_kp   = out + (size_t)BATCH * CHN * HW;         //      8,192
    float* out_zeta = out_kp + (size_t)BATCH * CHN * 2;       //      4,096

    float* gz = (float*)d_ws;                 // [B*C] zeta accumulators
    float* gx = gz + BATCH * CHN;             // [B*C] kpx accumulators
    float* gy = gx + BATCH * CHN;             // [B*C] kpy accumulators

    const int nacc = 3 * BATCH * CHN;
    init_acc_kernel<<<(nacc + 255) / 256, 256, 0, stream>>>(gz, nacc);

    dim3 grid(HW / TPB, BATCH);               // (36, 32)
    softmax_kp_kernel<<<grid, TPB, 0, stream>>>(in, out_map, gz, gx, gy);

    finalize_kernel<<<(BATCH * CHN + 255) / 256, 256, 0, stream>>>(
        gz, gx, gy, out_kp, out_zeta);
}